// Model_i2s_62182536511790
// MI455X (gfx1250) — compile-verified
//
#include <hip/hip_runtime.h>
#include <cstddef>

#define H 128
#define SLOPE 0.2f

static const int NI = 200000;   // items
static const int NS = 20000;    // sellers
static const int EE = 500000;   // edges per direction
static const int LL = 2;        // layers

typedef __attribute__((ext_vector_type(2))) float v2f;
typedef __attribute__((ext_vector_type(8))) float v8f;

// ---------------------------------------------------------------------------
// float atomic max via int punning (global_atomic_max_i32 / min_u32)
// ---------------------------------------------------------------------------
__device__ __forceinline__ void atomicMaxF(float* addr, float val) {
  if (val >= 0.0f) {
    atomicMax(reinterpret_cast<int*>(addr), __float_as_int(val));
  } else {
    atomicMin(reinterpret_cast<unsigned int*>(addr),
              static_cast<unsigned int>(__float_as_int(val)));
  }
}

// ---------------------------------------------------------------------------
// w_e[i] = sum_j W_edge[i][j] * a_edge[j]     (128x128 matvec, one block)
// ---------------------------------------------------------------------------
__global__ __launch_bounds__(128) void we_kernel(const float* __restrict__ W_edge,
                                                 const float* __restrict__ a_edge,
                                                 float* __restrict__ we) {
  int i = threadIdx.x;
  float s = 0.0f;
#pragma unroll 4
  for (int j = 0; j < H; ++j) s += W_edge[i * H + j] * a_edge[j];
  we[i] = s;
}

// ---------------------------------------------------------------------------
// Y = X @ W  (NxH @ HxH), fused rowdot[n] = sum_j Y[n][j]*avec[j]
// Block = 256 threads (8 wave32). Each block: 16 rows x 128 cols.
// Wave w owns the 16x16 N-tile at columns [16w, 16w+16).
// K-loop in steps of 4 using V_WMMA_F32_16X16X4_F32.
// ---------------------------------------------------------------------------
__global__ __launch_bounds__(256) void gemm_rowdot_kernel(
    const float* __restrict__ X, const float* __restrict__ W,
    const float* __restrict__ avec, float* __restrict__ Y,
    float* __restrict__ rowdot, int nrows) {
  __shared__ float As[16 * 132];  // padded stride to avoid LDS bank conflicts
  __shared__ float rsum[16];

  const int tid  = threadIdx.x;
  const int wave = tid >> 5;
  const int lane = tid & 31;
  const int half = lane >> 4;   // 0: lanes 0-15, 1: lanes 16-31
  const int l15  = lane & 15;
  const int r0   = blockIdx.x * 16;

  // Cooperative load of the 16x128 A tile into LDS (8 floats/thread).
  for (int i = tid; i < 16 * H; i += 256) {
    int r = i >> 7, c = i & (H - 1);
    int gr = r0 + r;
    As[r * 132 + c] = (gr < nrows) ? X[(size_t)gr * H + c] : 0.0f;
  }
  if (tid < 16) rsum[tid] = 0.0f;
  __syncthreads();

  const int colbase = wave * 16;
  v8f acc = {};

#pragma unroll 4
  for (int k = 0; k < H; k += 4) {
    // A frag (16x4): lanes 0-15 -> K=k,k+1 ; lanes 16-31 -> K=k+2,k+3
    v2f a, b;
    a[0] = As[l15 * 132 + k + 2 * half + 0];
    a[1] = As[l15 * 132 + k + 2 * half + 1];
    // B frag (4x16): W[k + {0,1 | 2,3}][colbase + l15]
    b[0] = W[(size_t)(k + 2 * half + 0) * H + colbase + l15];
    b[1] = W[(size_t)(k + 2 * half + 1) * H + colbase + l15];
    acc = __builtin_amdgcn_wmma_f32_16x16x4_f32(
        /*neg_a=*/false, a, /*neg_b=*/false, b,
        /*c_mod=*/(short)0, acc, /*reuse_a=*/false, /*reuse_b=*/false);
  }

  // D layout: VGPR i -> (lanes 0-15: M=i, lanes 16-31: M=8+i), N = colbase+l15
  const float aval = avec[colbase + l15];
#pragma unroll
  for (int i = 0; i < 8; ++i) {
    int row = i + 8 * half;
    int gr  = r0 + row;
    if (gr < nrows) Y[(size_t)gr * H + colbase + l15] = acc[i];
    float p = acc[i] * aval;
    // reduce the 16 lanes of each half-wave
    p += __shfl_xor(p, 1);
    p += __shfl_xor(p, 2);
    p += __shfl_xor(p, 4);
    p += __shfl_xor(p, 8);
    if (l15 == 0) atomicAdd(&rsum[row], p);  // ds_add_f32
  }
  __syncthreads();
  if (tid < 16 && (r0 + tid) < nrows) rowdot[r0 + tid] = rsum[tid];
}

// ---------------------------------------------------------------------------
// zero accumulator, init segment-max / denom
// ---------------------------------------------------------------------------
__global__ __launch_bounds__(256) void init_dst_kernel(float* __restrict__ accum,
                                                       float* __restrict__ m,
                                                       float* __restrict__ denom,
                                                       int ndst) {
  size_t i = (size_t)blockIdx.x * blockDim.x + threadIdx.x;
  if (i < (size_t)ndst * H) accum[i] = 0.0f;
  if (i < (size_t)ndst) {
    m[i] = -3.0e38f;
    denom[i] = 0.0f;
  }
}

// ---------------------------------------------------------------------------
// per-edge: a_e = <edge_attr[e], w_e>; logit = lrelu(a_s[src]+a_d[dst]+a_e)
// fused segment-max via atomicMaxF.  One wave32 per edge, float4 loads.
// ---------------------------------------------------------------------------
__global__ __launch_bounds__(256) void edge_logits_kernel(
    const int* __restrict__ src, const int* __restrict__ dst,
    const float* __restrict__ edge_attr, const float* __restrict__ we,
    const float* __restrict__ a_s, const float* __restrict__ a_d,
    float* __restrict__ logits, float* __restrict__ m, int nedges) {
  int e = (int)((blockIdx.x * (size_t)blockDim.x + threadIdx.x) >> 5);
  int lane = threadIdx.x & 31;
  if (e >= nedges) return;

  const float4* ea = reinterpret_cast<const float4*>(edge_attr + (size_t)e * H);
  const float4* wv = reinterpret_cast<const float4*>(we);
  if (e + 8 < nedges)
    __builtin_prefetch(edge_attr + (size_t)(e + 8) * H + lane * 4, 0, 1);

  float4 x = ea[lane];
  float4 w = wv[lane];
  float p = x.x * w.x + x.y * w.y + x.z * w.z + x.w * w.w;
  p += __shfl_xor(p, 1);
  p += __shfl_xor(p, 2);
  p += __shfl_xor(p, 4);
  p += __shfl_xor(p, 8);
  p += __shfl_xor(p, 16);
  if (lane == 0) {
    int d = dst[e];
    float logit = a_s[src[e]] + a_d[d] + p;
    float lr = logit > 0.0f ? logit : SLOPE * logit;
    logits[e] = lr;
    atomicMaxF(&m[d], lr);
  }
}

// ---------------------------------------------------------------------------
// ex = exp(logit - m[dst]); denom[dst] += ex  (logits overwritten with ex)
// ---------------------------------------------------------------------------
__global__ __launch_bounds__(256) void edge_exp_kernel(float* __restrict__ logits,
                                                       const int* __restrict__ dst,
                                                       const float* __restrict__ m,
                                                       float* __restrict__ denom,
                                                       int nedges) {
  int e = (int)(blockIdx.x * (size_t)blockDim.x + threadIdx.x);
  if (e >= nedges) return;
  int d = dst[e];
  float ex = __expf(logits[e] - m[d]);
  logits[e] = ex;
  atomicAdd(&denom[d], ex);
}

// ---------------------------------------------------------------------------
// out[dst] += (ex/denom[dst]) * xs[src]     (one wave32 per edge, float4)
// ---------------------------------------------------------------------------
__global__ __launch_bounds__(256) void edge_scatter_kernel(
    const int* __restrict__ src, const int* __restrict__ dst,
    const float* __restrict__ exv, const float* __restrict__ denom,
    const float* __restrict__ xs, float* __restrict__ out, int nedges) {
  int e = (int)((blockIdx.x * (size_t)blockDim.x + threadIdx.x) >> 5);
  int lane = threadIdx.x & 31;
  if (e >= nedges) return;

  int s = src[e];
  int d = dst[e];
  float alpha = exv[e] / (denom[d] + 1e-16f);
  const float4* xr = reinterpret_cast<const float4*>(xs + (size_t)s * H);
  float4 v = xr[lane];
  float* o = out + (size_t)d * H + lane * 4;
  atomicAdd(o + 0, alpha * v.x);
  atomicAdd(o + 1, alpha * v.y);
  atomicAdd(o + 2, alpha * v.z);
  atomicAdd(o + 3, alpha * v.w);
}

// ---------------------------------------------------------------------------
// out = relu(out + xd + bias)     (in place over the accumulator buffer)
// ---------------------------------------------------------------------------
__global__ __launch_bounds__(256) void finalize_kernel(float* __restrict__ out,
                                                       const float* __restrict__ xd,
                                                       const float* __restrict__ bias,
                                                       int ndst) {
  size_t i = (size_t)blockIdx.x * blockDim.x + threadIdx.x;
  if (i >= (size_t)ndst * H) return;
  float v = out[i] + xd[i] + bias[i & (H - 1)];
  out[i] = v > 0.0f ? v : 0.0f;
}

// ---------------------------------------------------------------------------
// One GATConv (one direction, one layer)
// ---------------------------------------------------------------------------
static void run_gat(const float* x_src, const float* x_dst, int n_src, int n_dst,
                    const int* src, const int* dst, const float* edge_attr,
                    const float* W, const float* a_src, const float* a_dst,
                    const float* W_edge, const float* a_edge, const float* bias,
                    float* xs, float* xd, float* a_sv, float* a_dv,
                    float* m, float* denom, float* logits, float* we,
                    float* out, hipStream_t stream) {
  we_kernel<<<1, 128, 0, stream>>>(W_edge, a_edge, we);

  gemm_rowdot_kernel<<<(n_src + 15) / 16, 256, 0, stream>>>(x_src, W, a_src, xs, a_sv, n_src);
  gemm_rowdot_kernel<<<(n_dst + 15) / 16, 256, 0, stream>>>(x_dst, W, a_dst, xd, a_dv, n_dst);

  size_t tot = (size_t)n_dst * H;
  init_dst_kernel<<<(unsigned)((tot + 255) / 256), 256, 0, stream>>>(out, m, denom, n_dst);

  unsigned ewaves = (unsigned)(((size_t)EE * 32 + 255) / 256);
  edge_logits_kernel<<<ewaves, 256, 0, stream>>>(src, dst, edge_attr, we, a_sv, a_dv,
                                                 logits, m, EE);
  edge_exp_kernel<<<(EE + 255) / 256, 256, 0, stream>>>(logits, dst, m, denom, EE);
  edge_scatter_kernel<<<ewaves, 256, 0, stream>>>(src, dst, logits, denom, xs, out, EE);

  finalize_kernel<<<(unsigned)((tot + 255) / 256), 256, 0, stream>>>(out, xd, bias, n_dst);
}

// ---------------------------------------------------------------------------
extern "C" void kernel_launch(void* const* d_in, const int* in_sizes, int n_in,
                              void* d_out, int out_size, void* d_ws, size_t ws_size,
                              hipStream_t stream) {
  (void)in_sizes; (void)n_in; (void)out_size; (void)ws_size;

  const float* x_item0   = (const float*)d_in[0];
  const float* x_seller0 = (const float*)d_in[1];
  const int*   ei_i2s    = (const int*)d_in[2];
  const int*   ei_s2i    = (const int*)d_in[3];
  const float* ea_i2s    = (const float*)d_in[4];
  const float* ea_s2i    = (const float*)d_in[5];
  const float* W_src_i2s   = (const float*)d_in[6];
  const float* att_src_i2s = (const float*)d_in[7];
  const float* att_dst_i2s = (const float*)d_in[8];
  const float* W_edge_i2s  = (const float*)d_in[9];
  const float* att_edge_i2s= (const float*)d_in[10];
  const float* bias_i2s    = (const float*)d_in[11];
  const float* W_src_s2i   = (const float*)d_in[12];
  const float* att_src_s2i = (const float*)d_in[13];
  const float* att_dst_s2i = (const float*)d_in[14];
  const float* W_edge_s2i  = (const float*)d_in[15];
  const float* att_edge_s2i= (const float*)d_in[16];
  const float* bias_s2i    = (const float*)d_in[17];

  const int* src_i2s = ei_i2s;        // items
  const int* dst_i2s = ei_i2s + EE;   // sellers
  const int* src_s2i = ei_s2i;        // sellers
  const int* dst_s2i = ei_s2i + EE;   // items

  // ---- workspace layout (floats) -------------------------------------
  const size_t NIH = (size_t)NI * H;   // 25,600,000
  const size_t NSH = (size_t)NS * H;   //  2,560,000
  float* ws      = (float*)d_ws;
  float* xs      = ws;                 // NIH
  float* xd      = xs + NIH;           // NIH
  float* itemA   = xd + NIH;           // NIH   (layer-0 item output)
  float* sellerA = itemA + NIH;        // NSH   (layer-0 seller output)
  float* a_sv    = sellerA + NSH;      // NI
  float* a_dv    = a_sv + NI;          // NI
  float* m       = a_dv + NI;          // NI
  float* denom   = m + NI;             // NI
  float* logits  = denom + NI;         // EE
  float* we      = logits + EE;        // H

  float* out_item_final   = (float*)d_out;        // NIH
  float* out_seller_final = (float*)d_out + NIH;  // NSH

  for (int l = 0; l < LL; ++l) {
    const float* xi = (l == 0) ? x_item0   : itemA;
    const float* xsell = (l == 0) ? x_seller0 : sellerA;
    float* out_seller = (l == LL - 1) ? out_seller_final : sellerA;
    float* out_item   = (l == LL - 1) ? out_item_final   : itemA;

    // seller_new = GAT_i2s(items -> sellers)
    run_gat(xi, xsell, NI, NS, src_i2s, dst_i2s, ea_i2s,
            W_src_i2s + (size_t)l * H * H, att_src_i2s + (size_t)l * H,
            att_dst_i2s + (size_t)l * H, W_edge_i2s + (size_t)l * H * H,
            att_edge_i2s + (size_t)l * H, bias_i2s + (size_t)l * H,
            xs, xd, a_sv, a_dv, m, denom, logits, we, out_seller, stream);

    // item_new = GAT_s2i(sellers -> items)
    run_gat(xsell, xi, NS, NI, src_s2i, dst_s2i, ea_s2i,
            W_src_s2i + (size_t)l * H * H, att_src_s2i + (size_t)l * H,
            att_dst_s2i + (size_t)l * H, W_edge_s2i + (size_t)l * H * H,
            att_edge_s2i + (size_t)l * H, bias_s2i + (size_t)l * H,
            xs, xd, a_sv, a_dv, m, denom, logits, we, out_item, stream);
  }
}